// Model_67525475827950
// MI455X (gfx1250) — compile-verified
//
#include <hip/hip_runtime.h>
#include <hip/hip_bf16.h>
#include <math.h>

// Problem constants (from the reference)
#define B_   1024
#define S_   200
#define NT_  10000
#define D_   256
#define K_   512      // 2*D
#define NU_  100000
#define BETA 0.1f

typedef __attribute__((ext_vector_type(2))) float v2f;
typedef __attribute__((ext_vector_type(8))) float v8f;

// ---------------------------------------------------------------------------
// Kernel 1: per-batch fused pass.
//   - l2norm(user_emb[user_id[b]])
//   - loop s: gather event_emb / pop_encoding rows, write pop[b,s,:],
//     accumulate mean(e), mean(sign(e)), mean(pop)
//   - write user_embeddings[b, 0:512]
// grid = B, block = 256 (one thread per d)
// ---------------------------------------------------------------------------
__global__ __launch_bounds__(256) void embed_pop_kernel(
    const int* __restrict__ user_id, const int* __restrict__ event_type,
    const float* __restrict__ pop_encoding, const float* __restrict__ event_emb,
    const float* __restrict__ user_emb,
    float* __restrict__ ue_out,   // [B, 512]
    float* __restrict__ pop_out)  // [B, S, D]
{
    const int b = blockIdx.x;
    const int d = threadIdx.x;

    float u = user_emb[(long)user_id[b] * D_ + d];

    // block reduction of sum(u*u) : wave32 shuffle + LDS
    float ss = u * u;
    #pragma unroll
    for (int off = 16; off > 0; off >>= 1) ss += __shfl_xor(ss, off, 32);
    __shared__ float red[8];
    if ((threadIdx.x & 31) == 0) red[threadIdx.x >> 5] = ss;
    __syncthreads();
    float tot = 0.f;
    #pragma unroll
    for (int w = 0; w < 8; ++w) tot += red[w];
    const float un = u * rsqrtf(fmaxf(tot, 1e-12f));

    __shared__ int ets[S_];
    if (d < S_) ets[d] = event_type[b * S_ + d];
    __syncthreads();

    float se = 0.f, ssg = 0.f, sp = 0.f;
    float* prow = pop_out + (long)b * S_ * D_ + d;
    for (int s = 0; s < S_; ++s) {
        const int t = ets[s];
        const float e = event_emb[(long)t * D_ + d];
        const float p = (t != 0) ? pop_encoding[(long)t * D_ + d] : 0.f;
        prow[(long)s * D_] = p;
        se += e;
        ssg += (e > 0.f) ? 1.f : ((e < 0.f) ? -1.f : 0.f);
        sp += p;
    }
    const float inv = 1.f / (float)S_;
    ue_out[(long)b * K_ + d]       = se * inv + un * (ssg * inv);
    ue_out[(long)b * K_ + D_ + d]  = sp * inv * BETA;
}

// ---------------------------------------------------------------------------
// Kernel 2: adjacency gather. grid = B, block = 256.
// Consecutive lanes share row idx[i]; prefetch next grid-stride element
// (global_prefetch_b8) to overlap the scattered-4B-read latency.
// ---------------------------------------------------------------------------
__global__ __launch_bounds__(256) void adj_kernel(
    const int* __restrict__ event_type, const float* __restrict__ adjm,
    float* __restrict__ adj_out)
{
    const int b = blockIdx.x;
    __shared__ int idx[S_];
    __shared__ int valid;
    if (threadIdx.x == 0) valid = 0;
    __syncthreads();
    int any = 0;
    for (int s = threadIdx.x; s < S_; s += blockDim.x) {
        const int t = event_type[b * S_ + s];
        idx[s] = (t > 0) ? (t - 1) : 0;
        any |= (t > 0);
    }
    if (any) atomicOr(&valid, 1);
    __syncthreads();
    const float hv = valid ? 1.f : 0.f;

    float* out = adj_out + (long)b * S_ * S_;
    for (int k = threadIdx.x; k < S_ * S_; k += blockDim.x) {
        const int kn = k + blockDim.x;
        if (kn < S_ * S_) {
            const int i2 = kn / S_;
            const int j2 = kn - i2 * S_;
            __builtin_prefetch(&adjm[(long)idx[i2] * NT_ + idx[j2]], 0, 0);
        }
        const int i = k / S_;
        const int j = k - i * S_;
        out[k] = adjm[(long)idx[i] * NT_ + idx[j]] * hv;
    }
}

// ---------------------------------------------------------------------------
// Kernel 3: scores = ue[B,512] @ item[NT,512]^T via V_WMMA_F32_16X16X4_F32.
// item row n = concat(event_emb[n+1], pop_encoding[n+1]*0.1). The *0.1 is
// folded into the A panel's second half during the LDS stage, so both K-half
// loops are branch-free pure loads feeding WMMAs.
// A panel (16 x 512) in LDS, row stride 516 floats (516%64==4 -> no bank
// conflicts for the 16-lane stride-LSTRIDE fragment reads).
// grid = (B/16, ceil(625/8)), block = 256 = 8 waves, one 16x16 C-tile per wave.
// ---------------------------------------------------------------------------
__global__ __launch_bounds__(256) void scores_wmma_kernel(
    const float* __restrict__ ue, const float* __restrict__ event_emb,
    const float* __restrict__ pop_encoding, float* __restrict__ scores)
{
    constexpr int LSTRIDE = 516;
    __shared__ float a_lds[16 * LSTRIDE];

    const int m0 = blockIdx.x * 16;
    // cooperative, coalesced A-panel load; fold BETA into second K-half
    for (int i = threadIdx.x; i < 16 * K_; i += 256) {
        const int m = i >> 9;          // /512
        const int k = i & (K_ - 1);    // %512
        const float v = ue[(long)(m0 + m) * K_ + k];
        a_lds[m * LSTRIDE + k] = (k < D_) ? v : v * BETA;
    }
    __syncthreads();

    const int wave = threadIdx.x >> 5;
    const int lane = threadIdx.x & 31;
    const int ntile = blockIdx.y * 8 + wave;   // 625 tiles of 16 columns

    if (ntile < NT_ / 16) {                    // wave-uniform: EXEC stays all-1s
        const int ncol = ntile * 16 + (lane & 15);
        const int half = lane >> 4;            // lanes 16..31 hold K+2,K+3
        const int mrow = lane & 15;

        // per-lane fragment base pointers (k offset = half*2, step 4)
        const float* abase = a_lds + mrow * LSTRIDE + half * 2;
        const float* bbase0 = event_emb    + (long)(ncol + 1) * D_ + half * 2;
        const float* bbase1 = pop_encoding + (long)(ncol + 1) * D_ + half * 2;

        v8f acc = {};
        // K-half 1: item cols 0..255 come from event_emb
        #pragma unroll 8
        for (int kk = 0; kk < D_; kk += 4) {
            const v2f a = *(const v2f*)(abase + kk);
            const v2f bf = *(const v2f*)(bbase0 + kk);
            acc = __builtin_amdgcn_wmma_f32_16x16x4_f32(
                false, a, false, bf, (short)0, acc, false, false);
        }
        // K-half 2: item cols 256..511 come from pop_encoding (scale in A)
        #pragma unroll 8
        for (int kk = 0; kk < D_; kk += 4) {
            const v2f a = *(const v2f*)(abase + D_ + kk);
            const v2f bf = *(const v2f*)(bbase1 + kk);
            acc = __builtin_amdgcn_wmma_f32_16x16x4_f32(
                false, a, false, bf, (short)0, acc, false, false);
        }

        // C layout: VGPR r, lanes 0-15 -> M=r, lanes 16-31 -> M=r+8, N=lane&15
        float* out = scores + (long)m0 * NT_ + ntile * 16 + (lane & 15);
        const int mbase = half * 8;
        #pragma unroll
        for (int r = 0; r < 8; ++r)
            out[(long)(mbase + r) * NT_] = acc[r];
    }
}

// ---------------------------------------------------------------------------
// Kernel 4: per-row sum of squares of scores (for l2norm over NT). grid = B.
// ---------------------------------------------------------------------------
__global__ __launch_bounds__(256) void rowsumsq_kernel(
    const float* __restrict__ scores, float* __restrict__ ss_out)
{
    const int b = blockIdx.x;
    const float* row = scores + (long)b * NT_;
    float s = 0.f;
    for (int i = threadIdx.x; i < NT_; i += 256) { const float v = row[i]; s += v * v; }
    #pragma unroll
    for (int off = 16; off > 0; off >>= 1) s += __shfl_xor(s, off, 32);
    __shared__ float red[8];
    if ((threadIdx.x & 31) == 0) red[threadIdx.x >> 5] = s;
    __syncthreads();
    if (threadIdx.x == 0) {
        float t = 0.f;
        #pragma unroll
        for (int w = 0; w < 8; ++w) t += red[w];
        ss_out[b] = t;
    }
}

// ---------------------------------------------------------------------------
// Kernel 5: prediction = tanh(score * rsqrt(max(ss, 1e-5))), in place.
// ---------------------------------------------------------------------------
__global__ __launch_bounds__(256) void finalize_kernel(
    const float* __restrict__ ss, float* __restrict__ scores)
{
    const long i = (long)blockIdx.x * blockDim.x + threadIdx.x;
    if (i < (long)B_ * NT_) {
        const int b = (int)(i / NT_);
        const float scale = rsqrtf(fmaxf(ss[b], 1e-5f));
        scores[i] = tanhf(scores[i] * scale);
    }
}

// ---------------------------------------------------------------------------
extern "C" void kernel_launch(void* const* d_in, const int* in_sizes, int n_in,
                              void* d_out, int out_size, void* d_ws, size_t ws_size,
                              hipStream_t stream)
{
    const int*   user_id      = (const int*)  d_in[0];
    const int*   event_type   = (const int*)  d_in[1];
    const float* adjacent     = (const float*)d_in[2];
    const float* pop_encoding = (const float*)d_in[3];
    // d_in[4] = evaluation scalar (0) -> non-eval path implemented
    const float* event_emb    = (const float*)d_in[5];
    const float* user_emb     = (const float*)d_in[6];

    float* out = (float*)d_out;
    // output layout (flat, return order): prediction, user_embeddings, pop, adj
    float* pred = out;                                   // B*NT
    float* ue   = pred + (long)B_ * NT_;                 // B*2D
    float* pop  = ue   + (long)B_ * K_;                  // B*S*D
    float* adj  = pop  + (long)B_ * S_ * D_;             // B*S*S

    float* ss = (float*)d_ws;                            // B floats of scratch

    // 1) fused gather/reduce: pop + user_embeddings
    embed_pop_kernel<<<B_, 256, 0, stream>>>(user_id, event_type, pop_encoding,
                                             event_emb, user_emb, ue, pop);
    // 2) adjacency gather
    adj_kernel<<<B_, 256, 0, stream>>>(event_type, adjacent, adj);
    // 3) WMMA GEMM -> raw scores into prediction region
    {
        dim3 grid(B_ / 16, (NT_ / 16 + 7) / 8);          // (64, 79)
        scores_wmma_kernel<<<grid, 256, 0, stream>>>(ue, event_emb, pop_encoding, pred);
    }
    // 4) row sum-of-squares for l2norm across NT
    rowsumsq_kernel<<<B_, 256, 0, stream>>>(pred, ss);
    // 5) tanh(l2norm(scores)) in place
    {
        const long total = (long)B_ * NT_;
        const int blocks = (int)((total + 255) / 256);
        finalize_kernel<<<blocks, 256, 0, stream>>>(ss, pred);
    }
}